// LearnableCurvatureAttention_43052752175154
// MI455X (gfx1250) — compile-verified
//
#include <hip/hip_runtime.h>
#include <math.h>

typedef __attribute__((ext_vector_type(2))) float v2f;
typedef __attribute__((ext_vector_type(8))) float v8f;

#define BB 4
#define SS 2048
#define DD 512
#define THRESH 0.01f
#define NEG_INF (-__builtin_inff())

#define WMMA_F32(A, Bv, C) \
    __builtin_amdgcn_wmma_f32_16x16x4_f32(false, (A), false, (Bv), (short)0, (C), false, false)

// ---------------------------------------------------------------------------
// Kernel 1: fused QKV projection, 32x16 tile per wave (2 row tiles x q/k/v).
// Per k-step: 5 b64 loads feed 6 WMMAs.  V is written TRANSPOSED [B,D,S] so
// the attn@V kernel gets contiguous K-major B fragments (2x float4 stores,
// exploiting the 8-consecutive-rows-per-lane C layout).
// ---------------------------------------------------------------------------
__global__ __launch_bounds__(128) void lca_qkv_kernel(
    const float* __restrict__ x,
    const float* __restrict__ Wq, const float* __restrict__ bq,
    const float* __restrict__ Wk, const float* __restrict__ bk,
    const float* __restrict__ Wv, const float* __restrict__ bv,
    float* __restrict__ qbuf, float* __restrict__ kbuf, float* __restrict__ vT)
{
    const int wave = blockIdx.x * 4 + (threadIdx.x >> 5);
    const int lane = threadIdx.x & 31;
    const int rt = wave >> 5;              // 0..255: 32-row blocks over B*S
    const int ct = wave & 31;              // 0..31 : 16-col blocks over D
    const int m  = lane & 15;
    const int kh = (lane >> 4) * 2;        // K sub-offset 0 / 2

    const float* xr0 = x  + (size_t)(rt * 32 + m) * DD + kh;
    const float* xr1 = xr0 + (size_t)16 * DD;
    const float* wqr = Wq + (size_t)(ct * 16 + m) * DD + kh;
    const float* wkr = Wk + (size_t)(ct * 16 + m) * DD + kh;
    const float* wvr = Wv + (size_t)(ct * 16 + m) * DD + kh;

    v8f cq0 = {}, cq1 = {}, ck0 = {}, ck1 = {}, cv0 = {}, cv1 = {};
#pragma unroll 4
    for (int kb = 0; kb < DD; kb += 4) {
        v2f a0 = *(const v2f*)(xr0 + kb);
        v2f a1 = *(const v2f*)(xr1 + kb);
        v2f bq_ = *(const v2f*)(wqr + kb);
        v2f bk_ = *(const v2f*)(wkr + kb);
        v2f bv_ = *(const v2f*)(wvr + kb);
        cq0 = WMMA_F32(a0, bq_, cq0);  cq1 = WMMA_F32(a1, bq_, cq1);
        ck0 = WMMA_F32(a0, bk_, ck0);  ck1 = WMMA_F32(a1, bk_, ck1);
        cv0 = WMMA_F32(a0, bv_, cv0);  cv1 = WMMA_F32(a1, bv_, cv1);
    }

    const int n  = ct * 16 + m;
    const int mb = (lane >> 4) * 8;
    const float biasq = bq[n], biask = bk[n], biasv = bv[n];

    auto store_tile = [&](int rowbase, const v8f& cq, const v8f& ck, const v8f& cv) {
#pragma unroll
        for (int r = 0; r < 8; ++r) {
            size_t off = (size_t)(rowbase + r) * DD + n;
            qbuf[off] = cq[r] + biasq;
            kbuf[off] = ck[r] + biask;
        }
        // transposed V store: 8 consecutive rows at fixed column n -> 2x float4
        const int bidx = rowbase >> 11;        // /S
        const int i    = rowbase & 2047;       // %S  (multiple of 8 -> 32B aligned)
        float* vp = vT + ((size_t)bidx * DD + n) * SS + i;
        float4 lo; lo.x = cv[0] + biasv; lo.y = cv[1] + biasv; lo.z = cv[2] + biasv; lo.w = cv[3] + biasv;
        float4 hi; hi.x = cv[4] + biasv; hi.y = cv[5] + biasv; hi.z = cv[6] + biasv; hi.w = cv[7] + biasv;
        *(float4*)(vp)     = lo;
        *(float4*)(vp + 4) = hi;
    };
    store_tile(rt * 32 + mb,      cq0, ck0, cv0);
    store_tile(rt * 32 + 16 + mb, cq1, ck1, cv1);
}

// ---------------------------------------------------------------------------
// Kernel 2: manifold projection (per-row scalar rescale) + squared norms.
// ---------------------------------------------------------------------------
__global__ __launch_bounds__(256) void lca_project_kernel(
    float* __restrict__ qbuf, float* __restrict__ kbuf,
    float* __restrict__ qn2, float* __restrict__ kn2,
    const float* __restrict__ curvature_raw)
{
    __shared__ float red[8];
    __shared__ float s_scale;
    const int which = blockIdx.x >> 13;            // B*S = 8192
    const int row   = blockIdx.x & 8191;
    float* buf   = which ? kbuf : qbuf;
    float* n2out = which ? kn2  : qn2;
    float* p = buf + (size_t)row * DD;

    float v0 = p[threadIdx.x];
    float v1 = p[threadIdx.x + 256];
    float ss = v0 * v0 + v1 * v1;
#pragma unroll
    for (int o = 16; o > 0; o >>= 1) ss += __shfl_down(ss, o, 32);
    if ((threadIdx.x & 31) == 0) red[threadIdx.x >> 5] = ss;
    __syncthreads();
    if (threadIdx.x == 0) {
        float sum = 0.f;
#pragma unroll
        for (int i = 0; i < 8; ++i) sum += red[i];
        const float kcur = 2.0f * tanhf(curvature_raw[0]);
        const float absk = fabsf(kcur);
        const float sk   = sqrtf(fmaxf(absk, 1e-5f));
        const float nrm  = sqrtf(sum + 1e-12f);
        float scale;
        if (absk < THRESH)      scale = 1.0f;
        else if (kcur < 0.0f)   scale = fminf(1.0f, ((1.0f - 1e-3f) / sk) / nrm);
        else                    scale = 1.0f / (nrm * sk);
        s_scale = scale;
        n2out[row] = scale * scale * sum;
    }
    __syncthreads();
    const float sc = s_scale;
    p[threadIdx.x]       = v0 * sc;
    p[threadIdx.x + 256] = v1 * sc;
}

// ---------------------------------------------------------------------------
// distance -> score transform (branch is wave-uniform on scalar curvature)
// ---------------------------------------------------------------------------
__device__ __forceinline__ float lca_score(float dot, float qi, float kj,
                                           float kcur, float absk, float sk,
                                           float invt)
{
    const float diff2 = fmaxf(qi + kj - 2.0f * dot, 0.0f);
    float dist;
    if (absk < THRESH) {
        dist = sqrtf(diff2 + 1e-12f);
    } else if (kcur < 0.0f) {
        const float denom = fmaxf((1.0f - absk * qi) * (1.0f - absk * kj), 1e-5f);
        const float arg   = 1.0f + 2.0f * absk * diff2 / denom;
        dist = acoshf(fmaxf(arg, 1.0f + 1e-7f)) / sk;
    } else {
        const float ca = fminf(fmaxf(absk * dot, -1.0f + 1e-7f), 1.0f - 1e-7f);
        dist = acosf(ca) / sk;
    }
    return -dist * invt;
}

// ---------------------------------------------------------------------------
// Kernel 3: curved-distance scores. 32x32 superblock per wave (2x2 WMMA tiles):
// per k-step 4 b64 loads feed 4 WMMAs.  Superblocks above the diagonal take a
// wave-uniform early exit (EXEC stays all-ones around WMMA).
// ---------------------------------------------------------------------------
__global__ __launch_bounds__(128) void lca_scores_kernel(
    const float* __restrict__ qm, const float* __restrict__ km,
    const float* __restrict__ qn2, const float* __restrict__ kn2,
    const float* __restrict__ curvature_raw, const float* __restrict__ temperature,
    float* __restrict__ attn)
{
    const int wave = blockIdx.x * 4 + (threadIdx.x >> 5);
    const int lane = threadIdx.x & 31;
    const int b    = wave >> 12;           // 64*64 superblocks per batch
    const int rem  = wave & 4095;
    const int it2  = rem >> 6;
    const int jt2  = rem & 63;
    const int n    = lane & 15;
    const int mb   = (lane >> 4) * 8;
    const int i0   = it2 * 32, j0 = jt2 * 32;
    const size_t bS = (size_t)b * SS;

    if (jt2 > it2) {   // fully masked superblock (wave-uniform)
#pragma unroll
        for (int ti = 0; ti < 2; ++ti)
#pragma unroll
            for (int r = 0; r < 8; ++r) {
                float* p = attn + (bS + i0 + ti * 16 + mb + r) * SS + j0 + n;
                p[0]  = NEG_INF;
                p[16] = NEG_INF;
            }
        return;
    }

    const int kh = (lane >> 4) * 2;
    const float* qr0 = qm + (bS + i0 + n) * DD + kh;
    const float* qr1 = qr0 + (size_t)16 * DD;
    const float* kr0 = km + (bS + j0 + n) * DD + kh;
    const float* kr1 = kr0 + (size_t)16 * DD;

    v8f c00 = {}, c01 = {}, c10 = {}, c11 = {};
#pragma unroll 4
    for (int kb = 0; kb < DD; kb += 4) {
        v2f a0 = *(const v2f*)(qr0 + kb);
        v2f a1 = *(const v2f*)(qr1 + kb);
        v2f b0 = *(const v2f*)(kr0 + kb);
        v2f b1 = *(const v2f*)(kr1 + kb);
        c00 = WMMA_F32(a0, b0, c00);  c01 = WMMA_F32(a0, b1, c01);
        c10 = WMMA_F32(a1, b0, c10);  c11 = WMMA_F32(a1, b1, c11);
    }

    const float kcur = 2.0f * tanhf(curvature_raw[0]);
    const float absk = fabsf(kcur);
    const float sk   = sqrtf(fmaxf(absk, 1e-5f));
    const float invt = 1.0f / (temperature[0] + 1e-8f);
    const float kj0  = kn2[bS + j0 + n];
    const float kj1  = kn2[bS + j0 + 16 + n];

    auto emit = [&](int ti, const v8f& cA, const v8f& cB) {
#pragma unroll
        for (int r = 0; r < 8; ++r) {
            const int i  = i0 + ti * 16 + mb + r;
            const float qi = qn2[bS + i];
            const int jA = j0 + n, jB = j0 + 16 + n;
            float sA = (jA > i) ? NEG_INF : lca_score(cA[r], qi, kj0, kcur, absk, sk, invt);
            float sB = (jB > i) ? NEG_INF : lca_score(cB[r], qi, kj1, kcur, absk, sk, invt);
            float* p = attn + (bS + i) * SS + j0 + n;
            p[0]  = sA;
            p[16] = sB;
        }
    };
    emit(0, c00, c01);
    emit(1, c10, c11);
}

// ---------------------------------------------------------------------------
// Kernel 4: row softmax in place. exp(-inf)=0 reproduces exact masked zeros.
// ---------------------------------------------------------------------------
__global__ __launch_bounds__(256) void lca_softmax_kernel(float* __restrict__ attn)
{
    __shared__ float red[8];
    __shared__ float s_max, s_sum;
    float* p = attn + (size_t)blockIdx.x * SS;

    float vals[8];
    float mx = NEG_INF;
#pragma unroll
    for (int e = 0; e < 8; ++e) {
        vals[e] = p[threadIdx.x + e * 256];
        mx = fmaxf(mx, vals[e]);
    }
#pragma unroll
    for (int o = 16; o > 0; o >>= 1) mx = fmaxf(mx, __shfl_down(mx, o, 32));
    if ((threadIdx.x & 31) == 0) red[threadIdx.x >> 5] = mx;
    __syncthreads();
    if (threadIdx.x == 0) {
        float m = red[0];
#pragma unroll
        for (int i = 1; i < 8; ++i) m = fmaxf(m, red[i]);
        s_max = m;
    }
    __syncthreads();
    const float m = s_max;

    float lsum = 0.f;
#pragma unroll
    for (int e = 0; e < 8; ++e) {
        vals[e] = expf(vals[e] - m);
        lsum += vals[e];
    }
#pragma unroll
    for (int o = 16; o > 0; o >>= 1) lsum += __shfl_down(lsum, o, 32);
    __syncthreads();
    if ((threadIdx.x & 31) == 0) red[threadIdx.x >> 5] = lsum;
    __syncthreads();
    if (threadIdx.x == 0) {
        float s = 0.f;
#pragma unroll
        for (int i = 0; i < 8; ++i) s += red[i];
        s_sum = s;
    }
    __syncthreads();
    const float inv = 1.0f / s_sum;
#pragma unroll
    for (int e = 0; e < 8; ++e)
        p[threadIdx.x + e * 256] = vals[e] * inv;
}

// ---------------------------------------------------------------------------
// Kernel 5: output = attn @ v using transposed vT[B,D,S]: all four fragments
// are contiguous b64 loads; 4 loads / 4 WMMAs per k-step.  K-loop truncated
// at i0+32 by causality (attn exactly 0 beyond).
// ---------------------------------------------------------------------------
__global__ __launch_bounds__(128) void lca_out_kernel(
    const float* __restrict__ attn, const float* __restrict__ vT,
    float* __restrict__ out)
{
    const int wave = blockIdx.x * 4 + (threadIdx.x >> 5);
    const int lane = threadIdx.x & 31;
    const int b    = wave >> 10;           // 64*16 superblocks per batch
    const int rem  = wave & 1023;
    const int it2  = rem >> 4;             // 0..63 over S/32
    const int dt2  = rem & 15;             // 0..15 over D/32
    const int n    = lane & 15;
    const int kh   = (lane >> 4) * 2;
    const int i0   = it2 * 32, d0 = dt2 * 32;
    const size_t bS = (size_t)b * SS;

    const float* ar0 = attn + (bS + i0 + n) * SS + kh;          // attn rows
    const float* ar1 = ar0 + (size_t)16 * SS;
    const float* vt0 = vT + ((size_t)b * DD + d0 + n) * SS + kh; // vT rows (= v cols)
    const float* vt1 = vt0 + (size_t)16 * SS;

    v8f c00 = {}, c01 = {}, c10 = {}, c11 = {};
    const int klim = i0 + 32;   // causal truncation
#pragma unroll 4
    for (int kb = 0; kb < klim; kb += 4) {
        v2f a0 = *(const v2f*)(ar0 + kb);
        v2f a1 = *(const v2f*)(ar1 + kb);
        v2f b0 = *(const v2f*)(vt0 + kb);
        v2f b1 = *(const v2f*)(vt1 + kb);
        c00 = WMMA_F32(a0, b0, c00);  c01 = WMMA_F32(a0, b1, c01);
        c10 = WMMA_F32(a1, b0, c10);  c11 = WMMA_F32(a1, b1, c11);
    }

    const int mb = (lane >> 4) * 8;
#pragma unroll
    for (int r = 0; r < 8; ++r) {
        float* p0 = out + (bS + i0 + mb + r) * DD + d0 + n;
        float* p1 = out + (bS + i0 + 16 + mb + r) * DD + d0 + n;
        p0[0]  = c00[r];
        p0[16] = c01[r];
        p1[0]  = c10[r];
        p1[16] = c11[r];
    }
}

// ---------------------------------------------------------------------------
extern "C" void kernel_launch(void* const* d_in, const int* in_sizes, int n_in,
                              void* d_out, int out_size, void* d_ws, size_t ws_size,
                              hipStream_t stream) {
    const float* x    = (const float*)d_in[0];
    const float* Wq   = (const float*)d_in[1];
    const float* bq   = (const float*)d_in[2];
    const float* Wk   = (const float*)d_in[3];
    const float* bk   = (const float*)d_in[4];
    const float* Wv   = (const float*)d_in[5];
    const float* bv   = (const float*)d_in[6];
    const float* craw = (const float*)d_in[7];
    const float* temp = (const float*)d_in[8];
    // d_in[9] = mask (causal triu(1)); recomputed analytically, not read.

    float* out  = (float*)d_out;                       // [B,S,D]
    float* attn = out + (size_t)BB * SS * DD;          // [B,S,S]

    float* ws   = (float*)d_ws;
    float* qbuf = ws;                                  // [B,S,D]
    float* kbuf = qbuf + (size_t)BB * SS * DD;         // [B,S,D]
    float* vT   = kbuf + (size_t)BB * SS * DD;         // [B,D,S]  (transposed V)
    float* qn2  = vT + (size_t)BB * SS * DD;           // [B,S]
    float* kn2  = qn2 + (size_t)BB * SS;               // [B,S]

    // 1) QKV: (B*S/32)*(D/16) = 8192 waves
    lca_qkv_kernel<<<2048, 128, 0, stream>>>(x, Wq, bq, Wk, bk, Wv, bv, qbuf, kbuf, vT);
    // 2) projection: 2*B*S rows
    lca_project_kernel<<<16384, 256, 0, stream>>>(qbuf, kbuf, qn2, kn2, craw);
    // 3) scores: B*(S/32)^2 = 16384 waves
    lca_scores_kernel<<<4096, 128, 0, stream>>>(qbuf, kbuf, qn2, kn2, craw, temp, attn);
    // 4) softmax: B*S rows
    lca_softmax_kernel<<<8192, 256, 0, stream>>>(attn);
    // 5) output GEMM: B*(S/32)*(D/32) = 4096 waves
    lca_out_kernel<<<1024, 128, 0, stream>>>(attn, vT, out);
}